// DifferentiableRobotModel_81767587381762
// MI455X (gfx1250) — compile-verified
//
#include <hip/hip_runtime.h>
#include <hip/hip_bf16.h>
#include <stdint.h>

#define NLINK 7
#define TILE 256
#define TILE_DW (TILE * NLINK)   // 1792 dwords = 7168 bytes per tile

typedef unsigned int u32x4 __attribute__((ext_vector_type(4)));
typedef int          i32x8 __attribute__((ext_vector_type(8)));
typedef int          i32x4 __attribute__((ext_vector_type(4)));

#if defined(__has_builtin)
#if __has_builtin(__builtin_amdgcn_tensor_load_to_lds)
#define HAVE_TDM 1
#endif
#endif
#ifndef HAVE_TDM
#define HAVE_TDM 0
#endif

// Issue an async DMA of `ndw` dwords from global `gsrc` into LDS `ldsDst`.
// TDM path: single Tensor-Data-Mover descriptor (1-D tile), issued by wave 0
// only (EXEC is ignored by TENSOR ops; TENSORcnt is per-wave).
__device__ __forceinline__ void fk_tile_fetch(float* ldsDst, const float* gsrc,
                                              int ndw, unsigned tid, unsigned bdim) {
#if HAVE_TDM
  if (tid < 32u) {
    unsigned long long ga = (unsigned long long)(uintptr_t)gsrc;
    unsigned lds_off = (unsigned)(uintptr_t)ldsDst;  // generic->LDS offset (addr[31:0])
    unsigned dim0 = (unsigned)ndw;

    // D# group 0: count=1 (valid), lds_addr[63:32], global_addr[120:64], type=2
    u32x4 g0;
    g0.x = 1u;
    g0.y = lds_off;
    g0.z = (unsigned)ga;
    g0.w = (unsigned)(ga >> 32) | (2u << 30);

    // D# group 1: wg_mask=0 (not in cluster), data_size=2 (4B),
    // tensor_dim0=ndw, tensor_dim1=1, tile_dim0=TILE_DW, tile_dim1=1,
    // tensor_dim0_stride=ndw.
    i32x8 g1;
    g1[0] = (int)(2u << 16);                                  // data_size=4B
    g1[1] = (int)((dim0 & 0xFFFFu) << 16);                    // tensor_dim0[15:0] @ bits 63:48
    g1[2] = (int)(((dim0 >> 16) & 0xFFFFu) | (1u << 16));     // dim0 hi @79:64, tensor_dim1=1 @95:80
    g1[3] = (int)((unsigned)TILE_DW << 16);                   // tile_dim0 @127:112
    g1[4] = 1;                                                // tile_dim1=1 @143:128
    g1[5] = (int)dim0;                                        // tensor_dim0_stride @191:160
    g1[6] = 0;
    g1[7] = 0;

    i32x4 z4 = {0, 0, 0, 0};                  // groups 2/3 unused (<=2D tensor)
    i32x8 z8 = {0, 0, 0, 0, 0, 0, 0, 0};      // 6-arg toolchain: extension block
    __builtin_amdgcn_tensor_load_to_lds(g0, g1, z4, z4, z8, 0);
  }
#else
  for (int j = (int)tid; j < ndw; j += (int)bdim) ldsDst[j] = gsrc[j];
#endif
}

__device__ __forceinline__ void fk_tile_wait(unsigned tid) {
#if HAVE_TDM
  if (tid < 32u) __builtin_amdgcn_s_wait_tensorcnt(0);
#endif
  __syncthreads();
}

__global__ __launch_bounds__(TILE) void fk7_kernel(const float* __restrict__ q,
                                                   const float* __restrict__ rf,   // [7][3][3]
                                                   const float* __restrict__ tfv,  // [7][3]
                                                   float* __restrict__ out,
                                                   int B, int numTiles) {
  __shared__ float sq[2][TILE_DW];
  const unsigned tid = threadIdx.x;
  const int t0 = (int)blockIdx.x;
  const int stride = (int)gridDim.x;
  if (t0 >= numTiles) return;

  const long long totalDW = (long long)B * NLINK;

  {  // prologue: kick off DMA of this block's first tile
    long long base = (long long)t0 * TILE_DW;
    long long rem = totalDW - base;
    int ndw = rem < (long long)TILE_DW ? (int)rem : TILE_DW;
    fk_tile_fetch(&sq[0][0], q + base, ndw, tid, blockDim.x);
  }

  int buf = 0;
  for (int t = t0; t < numTiles; t += stride) {
    fk_tile_wait(tid);  // tile t landed in sq[buf]; barrier publishes LDS

    int tn = t + stride;  // prefetch next tile into the other buffer
    if (tn < numTiles) {
      long long base = (long long)tn * TILE_DW;
      long long rem = totalDW - base;
      int ndw = rem < (long long)TILE_DW ? (int)rem : TILE_DW;
      fk_tile_fetch(&sq[buf ^ 1][0], q + base, ndw, tid, blockDim.x);
    }

    const int b = t * TILE + (int)tid;
    if (b < B) {
      const float* qq = &sq[buf][tid * NLINK];  // stride-7 -> bank-conflict-free
      const size_t Bsz = (size_t)B;
      float* quatBase = out + Bsz * (size_t)(NLINK * 3);

      float R00, R01, R02, R10, R11, R12, R20, R21, R22;
      float tx, ty, tz;

#pragma unroll
      for (int i = 0; i < NLINK; ++i) {
        const float qi = qq[i];
        const float c = __cosf(qi);
        const float s = __sinf(qi);

        // fixed-rotation rows (wave-uniform -> SMEM scalar loads)
        const float f00 = rf[i * 9 + 0], f01 = rf[i * 9 + 1], f02 = rf[i * 9 + 2];
        const float f10 = rf[i * 9 + 3], f11 = rf[i * 9 + 4], f12 = rf[i * 9 + 5];
        const float f20 = rf[i * 9 + 6], f21 = rf[i * 9 + 7], f22 = rf[i * 9 + 8];

        // Rj = Rf * Rot_axis(q), axes alternate z,y,z,y,z,y,z (constant-folded)
        float J00, J01, J02, J10, J11, J12, J20, J21, J22;
        if ((i & 1) == 0) {  // z-axis: cols (c*f0+s*f1, -s*f0+c*f1, f2)
          J00 = fmaf(c, f00, s * f01);  J01 = fmaf(c, f01, -s * f00);  J02 = f02;
          J10 = fmaf(c, f10, s * f11);  J11 = fmaf(c, f11, -s * f10);  J12 = f12;
          J20 = fmaf(c, f20, s * f21);  J21 = fmaf(c, f21, -s * f20);  J22 = f22;
        } else {             // y-axis: cols (c*f0-s*f2, f1, s*f0+c*f2)
          J00 = fmaf(c, f00, -s * f02); J01 = f01; J02 = fmaf(c, f02, s * f00);
          J10 = fmaf(c, f10, -s * f12); J11 = f11; J12 = fmaf(c, f12, s * f10);
          J20 = fmaf(c, f20, -s * f22); J21 = f21; J22 = fmaf(c, f22, s * f20);
        }

        const float g0 = tfv[i * 3 + 0], g1 = tfv[i * 3 + 1], g2 = tfv[i * 3 + 2];
        if (i == 0) {
          tx = g0; ty = g1; tz = g2;  // Rp = I, tp = 0
          R00 = J00; R01 = J01; R02 = J02;
          R10 = J10; R11 = J11; R12 = J12;
          R20 = J20; R21 = J21; R22 = J22;
        } else {
          // t = Rp*tf + tp  (uses parent rotation, before R update)
          tx = fmaf(R00, g0, fmaf(R01, g1, fmaf(R02, g2, tx)));
          ty = fmaf(R10, g0, fmaf(R11, g1, fmaf(R12, g2, ty)));
          tz = fmaf(R20, g0, fmaf(R21, g1, fmaf(R22, g2, tz)));
          // R = Rp * Rj
          const float n00 = fmaf(R00, J00, fmaf(R01, J10, R02 * J20));
          const float n01 = fmaf(R00, J01, fmaf(R01, J11, R02 * J21));
          const float n02 = fmaf(R00, J02, fmaf(R01, J12, R02 * J22));
          const float n10 = fmaf(R10, J00, fmaf(R11, J10, R12 * J20));
          const float n11 = fmaf(R10, J01, fmaf(R11, J11, R12 * J21));
          const float n12 = fmaf(R10, J02, fmaf(R11, J12, R12 * J22));
          const float n20 = fmaf(R20, J00, fmaf(R21, J10, R22 * J20));
          const float n21 = fmaf(R20, J01, fmaf(R21, J11, R22 * J21));
          const float n22 = fmaf(R20, J02, fmaf(R21, J12, R22 * J22));
          R00 = n00; R01 = n01; R02 = n02;
          R10 = n10; R11 = n11; R12 = n12;
          R20 = n20; R21 = n21; R22 = n22;
        }

        // quaternion (x,y,z,w) with sign-from-off-diagonal, per reference
        const float eps = 1e-9f;
        const float qw = 0.5f * sqrtf(fmaxf(1.0f + R00 + R11 + R22, eps));
        float qx = 0.5f * sqrtf(fmaxf(1.0f + R00 - R11 - R22, eps));
        float qy = 0.5f * sqrtf(fmaxf(1.0f - R00 + R11 - R22, eps));
        float qz = 0.5f * sqrtf(fmaxf(1.0f - R00 - R11 + R22, eps));
        qx = (R21 - R12 < 0.0f) ? -qx : qx;
        qy = (R02 - R20 < 0.0f) ? -qy : qy;
        qz = (R10 - R01 < 0.0f) ? -qz : qz;

        const size_t eb = (size_t)i * Bsz + (size_t)b;
        float* tsp = out + eb * 3;
        tsp[0] = tx; tsp[1] = ty; tsp[2] = tz;

        float* qp = quatBase + eb * 4;
        if ((B & 3) == 0) {  // uniform branch: quat region 16B-aligned -> b128 store
          *(float4*)qp = make_float4(qx, qy, qz, qw);
        } else {
          qp[0] = qx; qp[1] = qy; qp[2] = qz; qp[3] = qw;
        }
      }
    }
    buf ^= 1;
  }
}

extern "C" void kernel_launch(void* const* d_in, const int* in_sizes, int n_in,
                              void* d_out, int out_size, void* d_ws, size_t ws_size,
                              hipStream_t stream) {
  const float* q   = (const float*)d_in[0];  // [B,7]
  const float* rf  = (const float*)d_in[1];  // [7,3,3]
  const float* tfv = (const float*)d_in[2];  // [7,3]
  float* out = (float*)d_out;                // ts [7,B,3] ++ quats [7,B,4]

  const int B = in_sizes[0] / NLINK;
  const int numTiles = (B + TILE - 1) / TILE;
  int grid = numTiles < 1024 ? numTiles : 1024;  // 2 tiles/block at B=524288 -> DMA/compute overlap
  if (grid < 1) grid = 1;

  hipLaunchKernelGGL(fk7_kernel, dim3(grid), dim3(TILE), 0, stream,
                     q, rf, tfv, out, B, numTiles);
}